// CreateOverlappingWindows_5377299054741
// MI455X (gfx1250) — compile-verified
//
#include <hip/hip_runtime.h>
#include <cstdint>

// CreateOverlappingWindows: out[b,t,w*26+c] = x[b, t+w-9, c] (zero-padded in t).
// Equivalent to: out row (b,t) = contiguous 494-float slice of x starting at
// (b*T + t - 9)*26, with zero masking at batch-time boundaries.
// Pure data movement: ~135 MB of traffic -> ~5.8 us at 23.3 TB/s.
// Strategy: stage compact input tile in LDS via global_load_async_to_lds,
// then replicate via overlapped-LDS-read async stores (global_store_async_from_lds).

#define BATCH   32
#define TDIM    2000
#define CDIM    26            // floats per frame (104 bytes)
#define CTX     9
#define WIN     19
#define OUTC    (WIN * CDIM)  // 494 floats per output row (1976 bytes)
#define TT      50            // time rows per workgroup tile
#define TILES   (TDIM / TT)   // 40
#define LROWS   (TT + 2 * CTX)        // 68 staged rows
#define LFLOATS (LROWS * CDIM)        // 1768 floats = 7072 B LDS
#define THREADS 256

__global__ __launch_bounds__(THREADS)
void overlap_windows_async_kernel(const float* __restrict__ x,
                                  float* __restrict__ out) {
    __shared__ float smem[LFLOATS];

    const int tid  = threadIdx.x;
    const int b    = blockIdx.x / TILES;
    const int tile = blockIdx.x % TILES;
    const int t0   = tile * TT;

    // ---- Phase 0: zero the staging buffer (covers batch-edge padding rows) ----
    for (int i = tid; i < LFLOATS; i += THREADS) smem[i] = 0.0f;
    __syncthreads();  // compiler inserts s_wait_dscnt + barrier

    // ---- Phase 1: async-load the valid input slab into LDS (b64 units) ----
    // LDS row r holds global time t0 - 9 + r. Valid rows: r in [rlo, rhi).
    int rlo = CTX - t0;            if (rlo < 0)      rlo = 0;
    int rhi = TDIM + CTX - t0;     if (rhi > LROWS)  rhi = LROWS;
    const int nload = (rhi - rlo) * (CDIM / 2);   // 13 b64 units per row

    const unsigned lds_base = (unsigned)(uintptr_t)smem;      // LDS byte offset
    const unsigned lds_lo   = lds_base + (unsigned)(rlo * CDIM * 4);
    const uint64_t gsrc     = (uint64_t)(uintptr_t)x +
        ((uint64_t)b * TDIM * CDIM + (uint64_t)(t0 - CTX + rlo) * CDIM) * 4ull;

    for (int u = tid; u < nload; u += THREADS) {
        unsigned lofs = lds_lo + (unsigned)(u * 8);   // 8B aligned (104 = 13*8)
        unsigned gofs = (unsigned)(u * 8);            // 8B aligned
        asm volatile("global_load_async_to_lds_b64 %0, %1, %2"
                     :: "v"(lofs), "v"(gofs), "s"(gsrc)
                     : "memory");
    }
    asm volatile("s_wait_asynccnt 0x0" ::: "memory");
    __syncthreads();   // all waves' staged data visible before any store reads LDS

    // ---- Phase 2: async-store output rows straight out of LDS ----
    // Output row 'row' (global time t0+row) = 247 b64 units; LDS source starts
    // at byte row*104 (overlapping reads across rows implement the windowing).
    const uint64_t gdst = (uint64_t)(uintptr_t)out +
        ((uint64_t)b * TDIM + (uint64_t)t0) * (uint64_t)OUTC * 4ull;

    for (int e = tid; e < TT * (OUTC / 2); e += THREADS) {   // 12350 b64 units
        int row = e / (OUTC / 2);                 // constant-divisor magic mul
        int j   = e - row * (OUTC / 2);
        unsigned lofs = lds_base + (unsigned)(row * (CDIM * 4) + j * 8);
        unsigned gofs = (unsigned)(row * (OUTC * 4) + j * 8); // 1976 = 247*8
        asm volatile("global_store_async_from_lds_b64 %0, %1, %2"
                     :: "v"(gofs), "v"(lofs), "s"(gdst)
                     : "memory");
    }
    asm volatile("s_wait_asynccnt 0x0" ::: "memory");
    // S_ENDPGM performs an implicit wait-idle as well.
}

extern "C" void kernel_launch(void* const* d_in, const int* in_sizes, int n_in,
                              void* d_out, int out_size, void* d_ws, size_t ws_size,
                              hipStream_t stream) {
    const float* x  = (const float*)d_in[0];   // [32, 2000, 26] f32
    float*      out = (float*)d_out;           // [32, 2000, 494] f32

    dim3 grid(BATCH * TILES);   // 1280 workgroups
    dim3 block(THREADS);        // 8 wave32 waves
    overlap_windows_async_kernel<<<grid, block, 0, stream>>>(x, out);
}